// EulerRNNCell_78761110274356
// MI455X (gfx1250) — compile-verified
//
#include <hip/hip_runtime.h>
#include <hip/hip_bf16.h>

typedef __attribute__((ext_vector_type(2))) float v2f;
typedef __attribute__((ext_vector_type(8))) float v8f;

#define NTRAJ   64
#define BATCH   512
#define NSTEPS  255          // T-1
#define NTOT    (NTRAJ * BATCH)   // 32768

// ---------------------------------------------------------------------------
// Stage 1: Euler-Maruyama rollout, one thread per trajectory.
// wvec layout: [n][t][2] float32, n = traj*BATCH + b (per reference reshape).
// Writes xT component-major: ws[comp*NTOT + n].
// ---------------------------------------------------------------------------
__global__ void genois_rollout(const float* __restrict__ inputs,
                               const float* __restrict__ bloch0,
                               const float* __restrict__ wvec,
                               float* __restrict__ xT)
{
    const int n = blockIdx.x * blockDim.x + threadIdx.x;   // 0..NTOT-1
    const int b = n & (BATCH - 1);
    const float omega = inputs[b] + 1e-8f;

    float x1 = bloch0[0];
    float x2 = bloch0[1];
    float x3 = bloch0[2];

    const float2* __restrict__ w = reinterpret_cast<const float2*>(wvec) + (long)n * NSTEPS;

    const float SD  = 0.0625f;               // sqrt(DT) = 2^-4
    const float KC  = 0.44497190922573976f;  // sqrt(GAMMA*ETA/2) = sqrt(0.198)
    const float DTc = 0.00390625f;           // 2^-8
    const float HG  = 0.55f;                 // 0.5*GAMMA

    #pragma unroll 1
    for (int t = 0; t < NSTEPS; ++t) {
        if ((t & 15) == 0) {
            // keep the per-trajectory stream ahead of the dependency chain
            __builtin_prefetch((const void*)(w + t + 32), 0, 0);
        }
        float2 wv = w[t];
        float d1 = SD * wv.x;
        float d2 = SD * wv.y;

        float ax = -HG * x1;
        float ay = -omega * x3 - HG * x2;
        float az =  omega * x2;

        float bx = KC * (-x1 * x3 * d1 + x2 * d2);
        float by = KC * (-x2 * x3 * d1 - x1 * d2);
        float bz = KC * ((1.0f - x3 * x3) * d1);

        x1 = x1 + DTc * ax + bx;
        x2 = x2 + DTc * ay + by;
        x3 = x3 + DTc * az + bz;
    }

    xT[0 * NTOT + n] = x1;
    xT[1 * NTOT + n] = x2;
    xT[2 * NTOT + n] = x3;
}

// ---------------------------------------------------------------------------
// Stage 2: mean over 64 trajectories via V_WMMA_F32_16X16X4_F32.
// ones[16x4] x B[4x16] accumulated over 16 chunks -> column sums of 64 values,
// replicated in every row of D. A == all-ones makes the result independent of
// the exact K<->lane mapping in the B layout; only N = lane%16 matters (same
// mapping as C/D). Each wave produces 16 (comp,b) sums; 96 waves total.
// ---------------------------------------------------------------------------
__global__ void traj_mean_wmma(const float* __restrict__ xT,
                               float* __restrict__ xm)
{
    const int wave = (blockIdx.x * blockDim.x + threadIdx.x) >> 5;  // 0..95
    const int lane = threadIdx.x & 31;
    const int comp = wave >> 5;            // 0..2
    const int b0   = (wave & 31) << 4;     // 0,16,...,496
    const int col  = lane & 15;
    const int half = lane >> 4;            // lanes 16-31 carry the other 2 K rows

    const float* __restrict__ base = xT + comp * NTOT + (b0 + col);

    v8f acc = {};
    v2f aone; aone[0] = 1.0f; aone[1] = 1.0f;

    #pragma unroll 4
    for (int c = 0; c < 16; ++c) {
        const int t0 = 4 * c + 2 * half;
        v2f bm;
        bm[0] = base[(t0    ) * BATCH];
        bm[1] = base[(t0 + 1) * BATCH];
        acc = __builtin_amdgcn_wmma_f32_16x16x4_f32(
                  /*neg_a=*/false, aone,
                  /*neg_b=*/false, bm,
                  /*c_mod=*/(short)0, acc,
                  /*reuse_a=*/false, /*reuse_b=*/false);
    }

    const float mean = acc[0] * (1.0f / (float)NTRAJ);  // every D row identical
    if (lane < 16) {
        xm[comp * BATCH + b0 + col] = mean;
    }
}

// ---------------------------------------------------------------------------
// Stage 3: probabilities (clipped), append input, and rho (complex64 as
// interleaved re/im floats). out layout: [512*7] floats then [512*4*2] floats.
// ---------------------------------------------------------------------------
__global__ void genois_finalize(const float* __restrict__ xm,
                                const float* __restrict__ inputs,
                                float* __restrict__ out)
{
    const int b = blockIdx.x * blockDim.x + threadIdx.x;
    if (b >= BATCH) return;

    const float x = xm[0 * BATCH + b];
    const float y = xm[1 * BATCH + b];
    const float z = xm[2 * BATCH + b];

    const float px = 0.5f * (x + 1.0f);
    const float py = 0.5f * (y + 1.0f);
    const float pz = 0.5f * (z + 1.0f);

    float p[6] = { px, 1.0f - px, py, 1.0f - py, pz, 1.0f - pz };

    float* o = out + b * 7;
    #pragma unroll
    for (int j = 0; j < 6; ++j)
        o[j] = fminf(fmaxf(p[j], 0.0f), 1.0f);
    o[6] = inputs[b];

    float* r = out + BATCH * 7 + b * 8;  // complex64 -> 2 floats each
    r[0] = 0.5f * (1.0f + z);  r[1] = 0.0f;       // (1+z)/2
    r[2] = 0.5f * x;           r[3] = -0.5f * y;  // (x - i y)/2
    r[4] = 0.5f * x;           r[5] = 0.5f * y;   // (x + i y)/2
    r[6] = 0.5f * (1.0f - z);  r[7] = 0.0f;       // (1-z)/2
}

// ---------------------------------------------------------------------------
extern "C" void kernel_launch(void* const* d_in, const int* in_sizes, int n_in,
                              void* d_out, int out_size, void* d_ws, size_t ws_size,
                              hipStream_t stream)
{
    const float* inputs = (const float*)d_in[0];   // [512,1]
    const float* bloch0 = (const float*)d_in[1];   // [3]
    const float* wvec   = (const float*)d_in[2];   // [32768,255,2]
    float*       out    = (float*)d_out;           // 7680 floats

    float* xT = (float*)d_ws;                      // 3*32768 floats (384 KB)
    float* xm = xT + 3 * NTOT;                     // 3*512 floats

    // Stage 1: 32768 threads, 256/block (8 wave32 waves per block)
    genois_rollout<<<NTOT / 256, 256, 0, stream>>>(inputs, bloch0, wvec, xT);

    // Stage 2: 96 waves = 12 blocks x 256 threads, EXEC all-ones per wave
    traj_mean_wmma<<<12, 256, 0, stream>>>(xT, xm);

    // Stage 3: 512 threads
    genois_finalize<<<2, 256, 0, stream>>>(xm, inputs, out);
}